// AxonUnit_35983236006444
// MI455X (gfx1250) — compile-verified
//
#include <hip/hip_runtime.h>
#include <math.h>

typedef float v2f __attribute__((ext_vector_type(2)));
typedef float v8f __attribute__((ext_vector_type(8)));

// Problem shape (matches reference): X[B,T,N] fp32
constexpr int Bb = 16;
constexpr int Tt = 256;
constexpr int Nn = 32768;
constexpr int CHUNK = 16;            // timesteps per WMMA block (M dim)
constexpr int WAVES_PER_BLOCK = 8;   // 256 threads = 8 wave32

// One wave handles a strip of 16 channels x 256 timesteps of one batch.
// Blocked scan as matmul: Y_blk(16x16) = L(16x16) @ X_blk(16x16) + p (x) carry,
// realized as 4 chained v_wmma_f32_16x16x4_f32 (K split 16 = 4x4).
__global__ __launch_bounds__(256) void axon_scan_wmma(
    const float* __restrict__ X, const float* __restrict__ wptr,
    float* __restrict__ out) {
  const int lane = threadIdx.x & 31;
  const int waveInBlock = threadIdx.x >> 5;
  const int gw = blockIdx.x * WAVES_PER_BLOCK + waveInBlock;

  constexpr int groupsPerBatch = Nn / 16;  // 2048
  const int b  = gw / groupsPerBatch;
  const int ng = gw % groupsPerBatch;
  const int n0 = ng * 16;

  // a = sigmoid(w), uniform across all lanes
  const float w = wptr[0];
  const float a = 1.0f / (1.0f + expf(-w));

  // powers a^0 .. a^16
  float p[17];
  p[0] = 1.0f;
#pragma unroll
  for (int k = 1; k <= 16; ++k) p[k] = p[k - 1] * a;

  const int col  = lane & 15;   // channel within group / matrix column
  const int half = lane >> 4;   // half-wave select

  // A fragments (lower-triangular scan matrix L, 16x4 f32 per fragment):
  // lane m in [0,16): VGPR0 = L[m][k0], VGPR1 = L[m][k0+1];
  // lanes 16..31 carry K+2 (ISA 16x4 f32 A layout).
  v2f Af[4];
#pragma unroll
  for (int f = 0; f < 4; ++f) {
    const int k0 = 4 * f + 2 * half;
    const int m  = col;
    Af[f].x = (m >= k0)     ? p[m - k0]     : 0.0f;
    Af[f].y = (m >= k0 + 1) ? p[m - k0 - 1] : 0.0f;
  }

  // Carry factors for accumulator init: C[row][n] = a^(row+1) * carry[n],
  // where VGPR r holds row = r + 8*half (ISA 16x16 f32 C/D layout).
  float cf[8];
#pragma unroll
  for (int r = 0; r < 8; ++r) cf[r] = p[r + 8 * half + 1];

  const size_t batchBase = (size_t)b * Tt * Nn;
  const float* __restrict__ Xb = X + batchBase + n0 + col;
  float* __restrict__ Ob = out + batchBase + n0 + col;

  float carry = 0.0f;

  for (int t0 = 0; t0 < Tt; t0 += CHUNK) {
    // C = p (outer) carry
    v8f acc;
#pragma unroll
    for (int r = 0; r < 8; ++r) acc[r] = cf[r] * carry;

    // Load B fragments (4x16 f32 each): VGPR0 rows {4f, 4f+2} across halves,
    // VGPR1 rows {4f+1, 4f+3}. Each half-wave load = 64B contiguous, aligned.
    v2f Bf[4];
#pragma unroll
    for (int f = 0; f < 4; ++f) {
      const int r0 = 4 * f + 2 * half;
      Bf[f].x = Xb[(size_t)(t0 + r0) * Nn];
      Bf[f].y = Xb[(size_t)(t0 + r0 + 1) * Nn];
    }

    // Y_blk = L @ X_blk + C  via 4 chained K=4 WMMAs
#pragma unroll
    for (int f = 0; f < 4; ++f) {
      acc = __builtin_amdgcn_wmma_f32_16x16x4_f32(
          /*neg_a=*/false, Af[f], /*neg_b=*/false, Bf[f],
          /*c_mod=*/(short)0, acc, /*reuse_a=*/false, /*reuse_b=*/false);
    }

    // Store D: VGPR r -> row r + 8*half
#pragma unroll
    for (int r = 0; r < 8; ++r) {
      Ob[(size_t)(t0 + r + 8 * half) * Nn] = acc[r];
    }

    // Next carry = Y[t0+15][col] = acc[7] of lane (col+16)  (M=15 lives in
    // VGPR7, lanes 16..31). Broadcast to both half-waves.
    carry = __shfl(acc[7], 16 + col, 32);
  }
}

extern "C" void kernel_launch(void* const* d_in, const int* in_sizes, int n_in,
                              void* d_out, int out_size, void* d_ws, size_t ws_size,
                              hipStream_t stream) {
  (void)in_sizes; (void)n_in; (void)out_size; (void)d_ws; (void)ws_size;
  const float* X = (const float*)d_in[0];
  const float* w = (const float*)d_in[1];
  float* out = (float*)d_out;

  const int totalWaves = Bb * (Nn / 16);               // 32768
  const int blocks = totalWaves / WAVES_PER_BLOCK;     // 4096
  axon_scan_wmma<<<blocks, 256, 0, stream>>>(X, w, out);
}